// Probe_46231027974422
// MI455X (gfx1250) — compile-verified
//
#include <hip/hip_runtime.h>
#include <stdint.h>

// Problem constants (reference: x[2048, 16384, 8] fp32, W[1,40], b[1])
#define T_DIM   16384
#define C_DIM   8
#define BLOCK   256
#define TILE_T  1024                       // timesteps per TDM tile
#define NTILES  (T_DIM / TILE_T)           // 16 tiles
#define TILE_ELEMS (TILE_T * C_DIM)        // 8192 floats = 32 KB
#define STEPS_PER_THREAD (TILE_T / BLOCK)  // 4 timesteps per thread per tile

typedef unsigned int u32x4 __attribute__((ext_vector_type(4)));
typedef int          i32x8 __attribute__((ext_vector_type(8)));
typedef int          i32x4 __attribute__((ext_vector_type(4)));
typedef float        v2f   __attribute__((ext_vector_type(2)));
typedef float        v8f   __attribute__((ext_vector_type(8)));

// Issue one 32 KB contiguous tile via the Tensor Data Mover.
// D# layout per CDNA5 ISA ch.8: group0 = control/lds/global addr, group1 =
// dims/strides, groups 2-3 unused (2D tile, zero-filled).
__device__ __forceinline__ void tdm_load_tile(uint64_t gaddr, uint32_t lds_addr)
{
    u32x4 g0;
    g0.x = 1u;                                      // count=1 (valid), user mode
    g0.y = lds_addr;                                // lds_addr[31:0]  (bits 63:32)
    g0.z = (uint32_t)(gaddr & 0xFFFFFFFFull);       // global_addr[31:0]
    g0.w = (uint32_t)((gaddr >> 32) & 0x1FFFFFFull) // global_addr[56:32]
         | 0x80000000u;                             // type=2 ("image") in [127:126]

    i32x8 g1;
    g1[0] = 0x00020000;                  // data_size=2 (4 bytes); no mask/pad/iterate
    g1[1] = (int)(TILE_ELEMS << 16);     // tensor_dim0[15:0] in bits 63:48
    g1[2] = 0x00010000;                  // tensor_dim0[31:16]=0, tensor_dim1=1
    g1[3] = (int)(TILE_ELEMS << 16);     // tensor_dim1 hi=0, tile_dim0=8192
    g1[4] = 1;                           // tile_dim1=1, tile_dim2=0
    g1[5] = TILE_ELEMS;                  // tensor_dim0_stride[31:0]
    g1[6] = (int)(TILE_ELEMS << 16);     // stride0[47:32]=0, stride1[15:0]
    g1[7] = 0;                           // stride1[47:16]=0

    i32x4 gz4 = {0, 0, 0, 0};                       // groups 2/3 unused (<=2D)
    i32x8 gz8 = {0, 0, 0, 0, 0, 0, 0, 0};           // extra operand (zero-filled)

    __builtin_amdgcn_tensor_load_to_lds(g0, g1, gz4, gz4, gz8, 0 /*cpol*/);
}

__global__ __launch_bounds__(BLOCK)
void stats_head_kernel(const float* __restrict__ x,
                       const float* __restrict__ W,
                       const float* __restrict__ bias,
                       float* __restrict__ out)
{
    // Double-buffered 32 KB staging tiles filled by the TDM: 64 KB total
    __shared__ __align__(16) float tile[2][TILE_ELEMS];
    __shared__ float red[8][40];   // per-wave partial stats
    __shared__ float fin[40];      // block-combined stats

    const int tid  = threadIdx.x;
    const int row  = blockIdx.x;
    const int lane = tid & 31;
    const int wave = tid >> 5;

    float s1[C_DIM], s2[C_DIM], s3[C_DIM], cnt[C_DIM], mx[C_DIM];
    #pragma unroll
    for (int c = 0; c < C_DIM; ++c) {
        s1[c] = 0.f; s2[c] = 0.f; s3[c] = 0.f; cnt[c] = 0.f;
        mx[c] = -3.402823466e38f;
    }

    const uint64_t gbase = (uint64_t)(uintptr_t)(x + (size_t)row * T_DIM * C_DIM);
    const uint32_t ldsbuf[2] = {
        (uint32_t)(uintptr_t)(&tile[0][0]),
        (uint32_t)(uintptr_t)(&tile[1][0])
    };

    // ---- TDM-driven double-buffered pipeline: wave 0 runs the DMA ----
    if (wave == 0) {
        tdm_load_tile(gbase, ldsbuf[0]);
    }

    #pragma unroll 1
    for (int k = 0; k < NTILES; ++k) {
        if (wave == 0) {
            if (k + 1 < NTILES) {
                tdm_load_tile(gbase + (uint64_t)(k + 1) * (TILE_ELEMS * 4),
                              ldsbuf[(k + 1) & 1]);
                __builtin_amdgcn_s_wait_tensorcnt(1);  // tile k has landed
            } else {
                __builtin_amdgcn_s_wait_tensorcnt(0);
            }
        }
        __syncthreads();  // tile k visible to all waves

        #pragma unroll
        for (int j = 0; j < STEPS_PER_THREAD; ++j) {
            const float4* p = reinterpret_cast<const float4*>(
                &tile[k & 1][(j * BLOCK + tid) * C_DIM]);
            const float4 a  = p[0];   // ds_load_b128
            const float4 b4 = p[1];   // ds_load_b128
            const float v[8] = {a.x, a.y, a.z, a.w, b4.x, b4.y, b4.z, b4.w};

            #pragma unroll
            for (int c = 0; c < C_DIM; ++c) {
                const float xv = v[c];
                const float x2 = xv * xv;
                s1[c] += xv;
                s2[c] += x2;
                s3[c] += x2 * xv;
                mx[c]  = fmaxf(mx[c], xv);
                cnt[c] += (xv > 0.f) ? 1.f : 0.f;
            }
        }
        __syncthreads();  // reads done before this buffer is refilled
    }

    // ---- Intra-wave tree reduction (wave32 shuffles) ----
    #pragma unroll
    for (int c = 0; c < C_DIM; ++c) {
        #pragma unroll
        for (int off = 16; off > 0; off >>= 1) {
            s1[c]  += __shfl_xor(s1[c],  off, 32);
            s2[c]  += __shfl_xor(s2[c],  off, 32);
            s3[c]  += __shfl_xor(s3[c],  off, 32);
            cnt[c] += __shfl_xor(cnt[c], off, 32);
            mx[c]   = fmaxf(mx[c], __shfl_xor(mx[c], off, 32));
        }
    }

    if (lane == 0) {
        #pragma unroll
        for (int c = 0; c < C_DIM; ++c) {
            red[wave][c * 5 + 0] = mx[c];
            red[wave][c * 5 + 1] = s1[c];
            red[wave][c * 5 + 2] = cnt[c];
            red[wave][c * 5 + 3] = s2[c];
            red[wave][c * 5 + 4] = s3[c];
        }
    }
    __syncthreads();

    // ---- Cross-wave combine on wave 0 via V_WMMA_F32_16X16X4_F32 ----
    // B = all-ones makes D[m,n] = sum_k A[m,k] (layout-robust in B).
    // A layout (ISA 7.12.2): A[m,k] -> lane = m + 16*(k>=2), VGPR = k&1.
    // Pre-add wave pairs so K=4 covers all 8 waves: comb[k] = red[k]+red[k+4].
    if (wave == 0) {
        const int half = lane >> 4;       // 0: K=0,1   1: K=2,3
        const int m    = lane & 15;
        const int k0   = half * 2;

        v2f ones; ones.x = 1.f; ones.y = 1.f;
        v8f zc = {0.f, 0.f, 0.f, 0.f, 0.f, 0.f, 0.f, 0.f};

        #pragma unroll
        for (int t = 0; t < 3; ++t) {     // 3 x 16 rows cover 40 stats
            const int stat = t * 16 + m;
            float a0 = 0.f, a1 = 0.f;
            if (stat < 40) {
                a0 = red[k0    ][stat] + red[k0 + 4][stat];
                a1 = red[k0 + 1][stat] + red[k0 + 5][stat];
            }
            v2f A; A.x = a0; A.y = a1;
            v8f d = __builtin_amdgcn_wmma_f32_16x16x4_f32(
                false, A, false, ones, (short)0, zc, false, false);
            // D layout: VGPR j = row M = 16t + 8*half + j, same value all n-lanes.
            if (m == 0) {
                #pragma unroll
                for (int j = 0; j < 8; ++j) {
                    const int s = t * 16 + half * 8 + j;
                    if (s < 40) fin[s] = d[j];
                }
            }
        }
        // Max slots (stat = c*5) are not sums: overwrite with true max-combine.
        if (lane < 8) {
            const int s = lane * 5;
            float v = red[0][s];
            #pragma unroll
            for (int w = 1; w < 8; ++w) v = fmaxf(v, red[w][s]);
            fin[s] = v;
        }
    }
    __syncthreads();

    // ---- Final stats + 40-element dot with W (once per block) ----
    if (tid == 0) {
        const float Tf = (float)T_DIM;
        float acc = 0.f;
        #pragma unroll
        for (int c = 0; c < C_DIM; ++c) {
            const float mxv = fin[c * 5 + 0];
            const float S1  = fin[c * 5 + 1];
            const float CN  = fin[c * 5 + 2];
            const float S2  = fin[c * 5 + 3];
            const float S3  = fin[c * 5 + 4];

            const float mean = S1 / Tf;
            const float ex2  = S2 / Tf;
            float var = (S2 - Tf * mean * mean) / (Tf - 1.f);
            var = fmaxf(var, 0.f);              // guard fp rounding
            float sd = sqrtf(var);
            const float m3 = S3 / Tf - 3.f * mean * ex2 + 2.f * mean * mean * mean;
            const float d  = sd + 1e-8f;
            float skew = m3 / (d * d * d);
            if (sd != sd)     sd = 0.f;         // nan_to_num
            if (skew != skew) skew = 0.f;

            acc += W[c * 5 + 0] * mxv
                 + W[c * 5 + 1] * mean
                 + W[c * 5 + 2] * CN
                 + W[c * 5 + 3] * sd
                 + W[c * 5 + 4] * skew;
        }
        out[row] = acc + bias[0];
    }
}

extern "C" void kernel_launch(void* const* d_in, const int* in_sizes, int n_in,
                              void* d_out, int out_size, void* d_ws, size_t ws_size,
                              hipStream_t stream) {
    const float* x    = (const float*)d_in[0];   // [2048,16384,8] fp32
    const float* W    = (const float*)d_in[1];   // [1,40] fp32
    const float* bias = (const float*)d_in[2];   // [1] fp32
    float* out        = (float*)d_out;           // [2048,1] fp32

    const int B = out_size;                      // 2048 rows
    hipLaunchKernelGGL(stats_head_kernel, dim3(B), dim3(BLOCK), 0, stream,
                       x, W, bias, out);
}